// RobustInverseCompositional_1709396983890
// MI455X (gfx1250) — compile-verified
//
#include <hip/hip_runtime.h>
#include <math.h>

#define Bn 8
#define Hh 384
#define Ww 384
#define Cc 3
#define NPar 6
#define DELTA_ 10
#define TOL_ 1.0e-3f
#define LAM_ 0.05f
#define MAX_ITER_ 12
#define RIDGE_ 1.0e-6f

#define NPIX (Hh*Ww)          // 147456
#define TPB 256
#define NBLK (NPIX/TPB)       // 576 blocks per image (exact)
#define PSTR 32               // padded partial-vector stride (27 used, rows 27..31 zero)

typedef __attribute__((ext_vector_type(2))) float v2f;
typedef __attribute__((ext_vector_type(8))) float v8f;

// ---------------------------------------------------------------------------
// f32 WMMA 16x16x4: D = A(16x4) * B(4x16) + C, exact-precision f32 matrix op.
// Used as a deterministic columnwise reducer with B == all-ones.
// ---------------------------------------------------------------------------
__device__ __forceinline__ v8f wmma4(v2f a, v2f b, v8f c) {
  return __builtin_amdgcn_wmma_f32_16x16x4_f32(false, a, false, b, (short)0, c,
                                               false, false);
}

// Catmull-Rom-style cubic kernel, a = -0.5 (matches reference _cubic)
__device__ __forceinline__ float cubw(float s) {
  s = fabsf(s);
  float s2 = s * s, s3 = s2 * s;
  float w1 = 1.5f * s3 - 2.5f * s2 + 1.0f;
  float w2 = -0.5f * s3 + 2.5f * s2 - 4.0f * s + 2.0f;
  return (s <= 1.0f) ? w1 : ((s < 2.0f) ? w2 : 0.0f);
}

__device__ __forceinline__ void bicubic3(const float* __restrict__ I2, int b,
                                         float xg, float yg, float* out3) {
  float x0 = floorf(xg), y0 = floorf(yg);
  float tx = xg - x0, ty = yg - y0;
  int xi = (int)x0, yi = (int)y0;
  float wx[4], wy[4];
#pragma unroll
  for (int j = 0; j < 4; ++j) {
    wx[j] = cubw(tx - (float)(j - 1));
    wy[j] = cubw(ty - (float)(j - 1));
  }
  float r0 = 0.f, r1 = 0.f, r2 = 0.f;
#pragma unroll
  for (int j = 0; j < 4; ++j) {
    int yy = yi + (j - 1);
    yy = yy < 0 ? 0 : (yy > Hh - 1 ? Hh - 1 : yy);
    const float* row = I2 + ((size_t)(b * Hh + yy) * Ww) * Cc;
#pragma unroll
    for (int k = 0; k < 4; ++k) {
      int xx = xi + (k - 1);
      xx = xx < 0 ? 0 : (xx > Ww - 1 ? Ww - 1 : xx);
      float w = wy[j] * wx[k];
      const float* s = row + xx * Cc;
      r0 += w * s[0];
      r1 += w * s[1];
      r2 += w * s[2];
    }
  }
  out3[0] = r0; out3[1] = r1; out3[2] = r2;
}

// ---------------------------------------------------------------------------
// init: p_cur <- p_in, done <- 0
// ---------------------------------------------------------------------------
__global__ void init_kernel(const float* __restrict__ p_in,
                            float* __restrict__ p_cur, int* __restrict__ done) {
  int b = threadIdx.x;
  if (b < Bn) {
#pragma unroll
    for (int i = 0; i < NPar; ++i) p_cur[b * NPar + i] = p_in[b * NPar + i];
    done[b] = 0;
  }
}

// ---------------------------------------------------------------------------
// Per-pixel accumulation of b (6) + Hessian upper triangle (21).
// Block partials reduced with f32 WMMA (wave 0), written to partials[b][blk][32].
// ---------------------------------------------------------------------------
__global__ void accum_kernel(const float* __restrict__ I1,
                             const float* __restrict__ I2,
                             const float* __restrict__ p_cur,
                             float* __restrict__ partials) {
  __shared__ float sm[TPB * PSTR];
  const int b = blockIdx.y;
  const int tid = threadIdx.x;
  const int pix = blockIdx.x * TPB + tid;
  const int py = pix / Ww;
  const int px = pix - py * Ww;

  const float q0 = p_cur[b * NPar + 0], q1 = p_cur[b * NPar + 1];
  const float q2 = p_cur[b * NPar + 2], q3 = p_cur[b * NPar + 3];
  const float q4 = p_cur[b * NPar + 4], q5 = p_cur[b * NPar + 5];
  const float X = (float)px, Y = (float)py;
  const float xg = (1.0f + q2) * X + q3 * Y + q0;
  const float yg = q4 * X + (1.0f + q5) * Y + q1;

  float acc[27];
#pragma unroll
  for (int i = 0; i < 27; ++i) acc[i] = 0.0f;

  const bool v1 = (px >= DELTA_) && (px < Ww - DELTA_) &&
                  (py >= DELTA_) && (py < Hh - DELTA_);
  const float gx = rintf(xg), gy = rintf(yg);
  const bool wm = (gx >= (float)DELTA_) && (gx <= (float)(Ww - DELTA_)) &&
                  (gy >= (float)DELTA_) && (gy <= (float)(Hh - DELTA_));

  if (v1 && wm) {
    float Iw[3];
    bicubic3(I2, b, xg, yg, Iw);
    const int base = ((b * Hh + py) * Ww + px) * Cc;
    const float invlam = 1.0f / LAM_;
#pragma unroll
    for (int c = 0; c < Cc; ++c) {
      const float i1v = I1[base + c];
      const float DI = Iw[c] - i1v;
      const float t = DI * invlam;
      const float w = 1.0f / sqrtf(1.0f + t * t);
      const float gxv = 0.5f * (I1[base + Cc + c] - I1[base - Cc + c]);
      const float gyv = 0.5f * (I1[base + Ww * Cc + c] - I1[base - Ww * Cc + c]);
      const float d0 = gxv, d1 = gyv, d2 = gxv * X, d3 = gxv * Y,
                  d4 = gyv * X, d5 = gyv * Y;
      const float wDI = w * DI;
      acc[0] += d0 * wDI; acc[1] += d1 * wDI; acc[2] += d2 * wDI;
      acc[3] += d3 * wDI; acc[4] += d4 * wDI; acc[5] += d5 * wDI;
      const float w0 = w * d0, w1 = w * d1, w2 = w * d2, w3 = w * d3,
                  w4 = w * d4;
      acc[6]  += w0 * d0; acc[7]  += w0 * d1; acc[8]  += w0 * d2;
      acc[9]  += w0 * d3; acc[10] += w0 * d4; acc[11] += w0 * d5;
      acc[12] += w1 * d1; acc[13] += w1 * d2; acc[14] += w1 * d3;
      acc[15] += w1 * d4; acc[16] += w1 * d5;
      acc[17] += w2 * d2; acc[18] += w2 * d3; acc[19] += w2 * d4;
      acc[20] += w2 * d5;
      acc[21] += w3 * d3; acc[22] += w3 * d4; acc[23] += w3 * d5;
      acc[24] += w4 * d4; acc[25] += w4 * d5;
      acc[26] += w * d5 * d5;
    }
  }

#pragma unroll
  for (int i = 0; i < 27; ++i) sm[tid * PSTR + i] = acc[i];
#pragma unroll
  for (int i = 27; i < PSTR; ++i) sm[tid * PSTR + i] = 0.0f;
  __syncthreads();

  // Wave 0 (full wave32, EXEC all-ones) reduces 256 rows with f32 WMMA.
  // A[m][k] = sm[kbase+k][m]; B = ones -> D[m][n] = sum_k A[m][k].
  if (tid < 32) {
    const int lane = tid;
    const int m = lane & 15;
    const int kh = (lane >> 4) << 1;  // lanes 0-15: K0/K1, lanes 16-31: K2/K3
    v8f c0 = {};
    v8f c1 = {};
    v2f ones = {1.0f, 1.0f};
    for (int k = 0; k < TPB; k += 4) {
      v2f a0, a1;
      a0.x = sm[(k + kh) * PSTR + m];
      a0.y = sm[(k + kh + 1) * PSTR + m];
      a1.x = sm[(k + kh) * PSTR + 16 + m];
      a1.y = sm[(k + kh + 1) * PSTR + 16 + m];
      c0 = wmma4(a0, ones, c0);
      c1 = wmma4(a1, ones, c1);
    }
    float* out = partials + ((size_t)b * NBLK + blockIdx.x) * PSTR;
    // C/D layout: VGPR r, lanes0-15 -> (M=r, N=lane); lanes16-31 -> (M=8+r).
    if (lane == 0) {
#pragma unroll
      for (int r = 0; r < 8; ++r) { out[r] = c0[r]; out[16 + r] = c1[r]; }
    } else if (lane == 16) {
#pragma unroll
      for (int r = 0; r < 8; ++r) { out[8 + r] = c0[r]; out[24 + r] = c1[r]; }
    }
  }
}

// ---------------------------------------------------------------------------
// WMMA reduction of nblk partial 32-vectors per image -> accum[b][32]
// ---------------------------------------------------------------------------
__global__ void reduce_kernel(const float* __restrict__ partials,
                              float* __restrict__ accum, int nblk) {
  const int b = blockIdx.x;
  const int lane = threadIdx.x;
  const int m = lane & 15;
  const int kh = (lane >> 4) << 1;
  const float* P = partials + (size_t)b * nblk * PSTR;
  v8f c0 = {};
  v8f c1 = {};
  v2f ones = {1.0f, 1.0f};
  for (int k = 0; k < nblk; k += 4) {
    v2f a0, a1;
    a0.x = P[(k + kh) * PSTR + m];
    a0.y = P[(k + kh + 1) * PSTR + m];
    a1.x = P[(k + kh) * PSTR + 16 + m];
    a1.y = P[(k + kh + 1) * PSTR + 16 + m];
    c0 = wmma4(a0, ones, c0);
    c1 = wmma4(a1, ones, c1);
  }
  float* out = accum + b * PSTR;
  if (lane == 0) {
#pragma unroll
    for (int r = 0; r < 8; ++r) { out[r] = c0[r]; out[16 + r] = c1[r]; }
  } else if (lane == 16) {
#pragma unroll
    for (int r = 0; r < 8; ++r) { out[8 + r] = c0[r]; out[24 + r] = c1[r]; }
  }
}

// ---------------------------------------------------------------------------
// 6x6 solve (partial-pivot GE), affine inverse + composition, done gating.
// ---------------------------------------------------------------------------
__global__ void solve_kernel(const float* __restrict__ accum,
                             float* __restrict__ p_cur, int* __restrict__ done) {
  int b = threadIdx.x;
  if (b >= Bn) return;
  const float* S = accum + b * PSTR;

  float A[6][7];
  {
    int t = 6;
    for (int m = 0; m < 6; ++m)
      for (int n = m; n < 6; ++n) { A[m][n] = S[t]; A[n][m] = S[t]; ++t; }
    for (int m = 0; m < 6; ++m) { A[m][m] += RIDGE_; A[m][6] = S[m]; }
  }
  // Gaussian elimination with partial pivoting
  for (int col = 0; col < 6; ++col) {
    int piv = col;
    float best = fabsf(A[col][col]);
    for (int r = col + 1; r < 6; ++r) {
      float v = fabsf(A[r][col]);
      if (v > best) { best = v; piv = r; }
    }
    if (piv != col) {
      for (int cc = col; cc < 7; ++cc) {
        float tmp = A[col][cc]; A[col][cc] = A[piv][cc]; A[piv][cc] = tmp;
      }
    }
    float inv = 1.0f / A[col][col];
    for (int r = col + 1; r < 6; ++r) {
      float f = A[r][col] * inv;
      for (int cc = col; cc < 7; ++cc) A[r][cc] -= f * A[col][cc];
    }
  }
  float dp[6];
  for (int m = 5; m >= 0; --m) {
    float s = A[m][6];
    for (int n = m + 1; n < 6; ++n) s -= A[m][n] * dp[n];
    dp[m] = s / A[m][m];
  }
  float nrm = sqrtf(dp[0]*dp[0] + dp[1]*dp[1] + dp[2]*dp[2] +
                    dp[3]*dp[3] + dp[4]*dp[4] + dp[5]*dp[5]);
  bool conv = nrm < TOL_;

  int was_done = done[b];
  if (!was_done) {
    // Mp from current params
    float pa = 1.0f + p_cur[b*NPar+2], pb = p_cur[b*NPar+3], ptx = p_cur[b*NPar+0];
    float pc = p_cur[b*NPar+4], pd = 1.0f + p_cur[b*NPar+5], pty = p_cur[b*NPar+1];
    // Mdp and its affine inverse
    float a2 = 1.0f + dp[2], b2 = dp[3], t2x = dp[0];
    float c2 = dp[4], d2 = 1.0f + dp[5], t2y = dp[1];
    float det = a2 * d2 - b2 * c2;
    float ia = d2 / det, ib = -b2 / det, ic = -c2 / det, id = a2 / det;
    float itx = -(ia * t2x + ib * t2y);
    float ity = -(ic * t2x + id * t2y);
    // Mn = Mp * inv(Mdp)
    float n00 = pa * ia + pb * ic;
    float n01 = pa * ib + pb * id;
    float n02 = pa * itx + pb * ity + ptx;
    float n10 = pc * ia + pd * ic;
    float n11 = pc * ib + pd * id;
    float n12 = pc * itx + pd * ity + pty;
    p_cur[b*NPar+0] = n02;
    p_cur[b*NPar+1] = n12;
    p_cur[b*NPar+2] = n00 - 1.0f;
    p_cur[b*NPar+3] = n01;
    p_cur[b*NPar+4] = n10;
    p_cur[b*NPar+5] = n11 - 1.0f;
  }
  if (conv) done[b] = 1;
}

// ---------------------------------------------------------------------------
// Final warp: write full Iw and masked DI; block-partial [rho_sum, count].
// ---------------------------------------------------------------------------
__global__ void final_kernel(const float* __restrict__ I1,
                             const float* __restrict__ I2,
                             const float* __restrict__ p_cur,
                             float* __restrict__ out_DI,
                             float* __restrict__ out_Iw,
                             float* __restrict__ partials) {
  __shared__ float sr[TPB];
  __shared__ float sc[TPB];
  const int b = blockIdx.y;
  const int tid = threadIdx.x;
  const int pix = blockIdx.x * TPB + tid;
  const int py = pix / Ww;
  const int px = pix - py * Ww;

  const float q0 = p_cur[b * NPar + 0], q1 = p_cur[b * NPar + 1];
  const float q2 = p_cur[b * NPar + 2], q3 = p_cur[b * NPar + 3];
  const float q4 = p_cur[b * NPar + 4], q5 = p_cur[b * NPar + 5];
  const float X = (float)px, Y = (float)py;
  const float xg = (1.0f + q2) * X + q3 * Y + q0;
  const float yg = q4 * X + (1.0f + q5) * Y + q1;

  const bool v1 = (px >= DELTA_) && (px < Ww - DELTA_) &&
                  (py >= DELTA_) && (py < Hh - DELTA_);
  const float gx = rintf(xg), gy = rintf(yg);
  const bool wm = (gx >= (float)DELTA_) && (gx <= (float)(Ww - DELTA_)) &&
                  (gy >= (float)DELTA_) && (gy <= (float)(Hh - DELTA_));
  const bool m = v1 && wm;

  float Iw[3];
  bicubic3(I2, b, xg, yg, Iw);
  const int base = ((b * Hh + py) * Ww + px) * Cc;

  float rs = 0.0f, cn = 0.0f;
#pragma unroll
  for (int c = 0; c < Cc; ++c) {
    out_Iw[base + c] = Iw[c];
    float DI = 0.0f;
    if (m) {
      DI = Iw[c] - I1[base + c];
      float t = DI * (1.0f / LAM_);
      rs += 2.0f * LAM_ * LAM_ * (sqrtf(1.0f + t * t) - 1.0f);
      cn += 1.0f;
    }
    out_DI[base + c] = DI;
  }

  sr[tid] = rs; sc[tid] = cn;
  __syncthreads();
  for (int s = TPB / 2; s > 0; s >>= 1) {
    if (tid < s) { sr[tid] += sr[tid + s]; sc[tid] += sc[tid + s]; }
    __syncthreads();
  }
  if (tid == 0) {
    float* o = partials + ((size_t)b * NBLK + blockIdx.x) * PSTR;
    o[0] = sr[0];
    o[1] = sc[0];
    for (int i = 2; i < PSTR; ++i) o[i] = 0.0f;
  }
}

__global__ void finalize_kernel(const float* __restrict__ accum,
                                const float* __restrict__ p_cur,
                                float* __restrict__ out_pf,
                                float* __restrict__ out_err) {
  int b = threadIdx.x;
  if (b < Bn) {
#pragma unroll
    for (int i = 0; i < NPar; ++i) out_pf[b * NPar + i] = p_cur[b * NPar + i];
    float cnt = accum[b * PSTR + 1];
    out_err[b] = accum[b * PSTR + 0] / fmaxf(cnt, 1.0f);
  }
}

// ---------------------------------------------------------------------------
extern "C" void kernel_launch(void* const* d_in, const int* in_sizes, int n_in,
                              void* d_out, int out_size, void* d_ws,
                              size_t ws_size, hipStream_t stream) {
  (void)in_sizes; (void)n_in; (void)out_size; (void)ws_size;
  const float* I1 = (const float*)d_in[0];
  const float* I2 = (const float*)d_in[1];
  const float* p  = (const float*)d_in[2];
  float* out = (float*)d_out;

  // workspace layout (floats): partials[B][NBLK][32] | accum[B][32] | p_cur[B][6] | done[B]
  float* ws = (float*)d_ws;
  float* partials = ws;
  float* accum = partials + (size_t)Bn * NBLK * PSTR;
  float* p_cur = accum + Bn * PSTR;
  int* done = (int*)(p_cur + Bn * NPar);

  init_kernel<<<1, 32, 0, stream>>>(p, p_cur, done);

  dim3 grid(NBLK, Bn);
  for (int it = 0; it < MAX_ITER_; ++it) {
    accum_kernel<<<grid, TPB, 0, stream>>>(I1, I2, p_cur, partials);
    reduce_kernel<<<Bn, 32, 0, stream>>>(partials, accum, NBLK);
    solve_kernel<<<1, 32, 0, stream>>>(accum, p_cur, done);
  }

  float* out_pf = out;                                   // [8,6]
  float* out_err = out + Bn * NPar;                      // [8]
  float* out_DI = out_err + Bn;                          // [8,384,384,3]
  float* out_Iw = out_DI + (size_t)Bn * Hh * Ww * Cc;    // [8,384,384,3]

  final_kernel<<<grid, TPB, 0, stream>>>(I1, I2, p_cur, out_DI, out_Iw, partials);
  reduce_kernel<<<Bn, 32, 0, stream>>>(partials, accum, NBLK);
  finalize_kernel<<<1, 32, 0, stream>>>(accum, p_cur, out_pf, out_err);
}